// RasterizeGaussians_4234837754027
// MI455X (gfx1250) — compile-verified
//
#include <hip/hip_runtime.h>
#include <hip/hip_bf16.h>
#include <stdint.h>

#define HH 256
#define WW 256
#define TILEPX 16
#define PTS 4096
#define CHUNKN 256
#define NCHUNK (PTS / CHUNKN)

typedef __attribute__((ext_vector_type(2))) float v2f;
typedef __attribute__((ext_vector_type(8))) float v8f;

// ---- CDNA5-specific helpers -------------------------------------------------

// Low 32 bits of a generic pointer to LDS == LDS byte offset (aperture layout).
__device__ __forceinline__ uint32_t lds_off_u32(const void* p) {
    return (uint32_t)(uintptr_t)p;
}

// Async global->LDS copy (ASYNCcnt path). VDST = LDS byte-address VGPR,
// VADDR = 64-bit global address VGPR pair, saddr = off.
__device__ __forceinline__ void async_copy_b128(uint32_t lds_off, const float* gsrc) {
    unsigned long long ga = (unsigned long long)(uintptr_t)gsrc;
    asm volatile("global_load_async_to_lds_b128 %0, %1, off"
                 :: "v"(lds_off), "v"(ga) : "memory");
}

__device__ __forceinline__ void wait_async0() {
    asm volatile("s_wait_asynccnt 0x0" ::: "memory");
}

// dst[lane] = v[srcLane]  (wave32: ds_bpermute uses byte index, bits [6:2])
__device__ __forceinline__ float lane_gather(float v, int srcLane) {
    return __int_as_float(__builtin_amdgcn_ds_bpermute(srcLane << 2, __float_as_int(v)));
}

// ---- K1: per-Gaussian preprocessing ----------------------------------------

__global__ __launch_bounds__(256) void gsplat_preprocess(
    const float* __restrict__ means, const float* __restrict__ sh,
    const float* __restrict__ opac, const float* __restrict__ scales,
    const float* __restrict__ rots, const float* __restrict__ vm,
    const float* __restrict__ pm, const float* __restrict__ campos,
    float* __restrict__ packed, float* __restrict__ out_radii,
    float* __restrict__ out_vis, float* __restrict__ out_xy)
{
    int g = blockIdx.x * 256 + threadIdx.x;
    if (g >= PTS) return;
    const float TANF = 0.5f;
    const float fx = WW / (2.f * TANF);
    const float fy = HH / (2.f * TANF);

    float m0 = means[3*g+0], m1 = means[3*g+1], m2 = means[3*g+2];

    float pv0 = vm[0]*m0 + vm[1]*m1 + vm[2]*m2  + vm[3];
    float pv1 = vm[4]*m0 + vm[5]*m1 + vm[6]*m2  + vm[7];
    float pv2 = vm[8]*m0 + vm[9]*m1 + vm[10]*m2 + vm[11];
    float h0  = pm[0]*m0  + pm[1]*m1  + pm[2]*m2  + pm[3];
    float h1  = pm[4]*m0  + pm[5]*m1  + pm[6]*m2  + pm[7];
    float h3  = pm[12]*m0 + pm[13]*m1 + pm[14]*m2 + pm[15];
    float invw = 1.f / (h3 + 1e-7f);
    float prx = h0 * invw, pry = h1 * invw;

    float depth = pv2;
    bool  in_front = depth > 0.2f;
    float tz = in_front ? depth : 1.f;

    // quaternion -> rotation
    float qr = rots[4*g+0], qx = rots[4*g+1], qy = rots[4*g+2], qz = rots[4*g+3];
    float qn = rsqrtf(qr*qr + qx*qx + qy*qy + qz*qz);
    qr *= qn; qx *= qn; qy *= qn; qz *= qn;
    float R00 = 1.f - 2.f*(qy*qy + qz*qz), R01 = 2.f*(qx*qy - qr*qz), R02 = 2.f*(qx*qz + qr*qy);
    float R10 = 2.f*(qx*qy + qr*qz), R11 = 1.f - 2.f*(qx*qx + qz*qz), R12 = 2.f*(qy*qz - qr*qx);
    float R20 = 2.f*(qx*qz - qr*qy), R21 = 2.f*(qy*qz + qr*qx), R22 = 1.f - 2.f*(qx*qx + qy*qy);

    float s0 = scales[3*g+0], s1 = scales[3*g+1], s2 = scales[3*g+2];
    float w0 = s0*s0, w1 = s1*s1, w2 = s2*s2;  // SCALE_MOD = 1
    // Sigma = R diag(w) R^T (symmetric)
    float S00 = R00*R00*w0 + R01*R01*w1 + R02*R02*w2;
    float S01 = R00*R10*w0 + R01*R11*w1 + R02*R12*w2;
    float S02 = R00*R20*w0 + R01*R21*w1 + R02*R22*w2;
    float S11 = R10*R10*w0 + R11*R11*w1 + R12*R12*w2;
    float S12 = R10*R20*w0 + R11*R21*w1 + R12*R22*w2;
    float S22 = R20*R20*w0 + R21*R21*w1 + R22*R22*w2;

    float limx = 1.3f * TANF, limy = 1.3f * TANF;
    float txv = fminf(fmaxf(pv0 / tz, -limx), limx) * tz;
    float tyv = fminf(fmaxf(pv1 / tz, -limy), limy) * tz;
    float iz = 1.f / tz, iz2 = iz * iz;
    float J00 = fx * iz, J02 = -fx * txv * iz2;
    float J11 = fy * iz, J12 = -fy * tyv * iz2;
    // T23 = J @ VM[:3,:3]
    float T00 = J00*vm[0] + J02*vm[8];
    float T01 = J00*vm[1] + J02*vm[9];
    float T02 = J00*vm[2] + J02*vm[10];
    float T10 = J11*vm[4] + J12*vm[8];
    float T11 = J11*vm[5] + J12*vm[9];
    float T12 = J11*vm[6] + J12*vm[10];
    // cov2 = T23 * Sigma * T23^T
    float U00 = T00*S00 + T01*S01 + T02*S02;
    float U01 = T00*S01 + T01*S11 + T02*S12;
    float U02 = T00*S02 + T01*S12 + T02*S22;
    float U10 = T10*S00 + T11*S01 + T12*S02;
    float U11 = T10*S01 + T11*S11 + T12*S12;
    float U12 = T10*S02 + T11*S12 + T12*S22;
    float aa = U00*T00 + U01*T01 + U02*T02 + 0.3f;
    float bb = U00*T10 + U01*T11 + U02*T12;
    float cc = U10*T10 + U11*T11 + U12*T12 + 0.3f;

    float det = aa*cc - bb*bb;
    bool  det_ok = det > 0.f;
    float inv_det = det_ok ? 1.f/det : 0.f;
    float conA =  cc*inv_det, conB = -bb*inv_det, conC = aa*inv_det;
    float mid = 0.5f*(aa + cc);
    float disc = sqrtf(fmaxf(0.1f, mid*mid - det));
    float radius = ceilf(3.f * sqrtf(fmaxf(mid + disc, mid - disc)));

    float px = ((prx + 1.f)*WW - 1.f)*0.5f;
    float py = ((pry + 1.f)*HH - 1.f)*0.5f;
    float rminx = fminf(fmaxf(floorf((px - radius)/16.f), 0.f), 16.f);
    float rminy = fminf(fmaxf(floorf((py - radius)/16.f), 0.f), 16.f);
    float rmaxx = fminf(fmaxf(floorf((px + radius + 15.f)/16.f), 0.f), 16.f);
    float rmaxy = fminf(fmaxf(floorf((py + radius + 15.f)/16.f), 0.f), 16.f);
    float tiles = (rmaxx - rminx)*(rmaxy - rminy);
    bool  visible = in_front && det_ok && (tiles > 0.f);
    int   radii = visible ? (int)radius : 0;

    // SH degree-3 eval
    float dx = m0 - campos[0], dy = m1 - campos[1], dz = m2 - campos[2];
    float dn = rsqrtf(dx*dx + dy*dy + dz*dz);
    dx *= dn; dy *= dn; dz *= dn;
    float xx = dx*dx, yy = dy*dy, zz = dz*dz, xy = dx*dy, yz = dy*dz, xz = dx*dz;
    float basis[16];
    basis[0]  = 0.28209479177387814f;
    basis[1]  = -0.4886025119029199f * dy;
    basis[2]  =  0.4886025119029199f * dz;
    basis[3]  = -0.4886025119029199f * dx;
    basis[4]  =  1.0925484305920792f * xy;
    basis[5]  = -1.0925484305920792f * yz;
    basis[6]  =  0.31539156525252005f * (2.f*zz - xx - yy);
    basis[7]  = -1.0925484305920792f * xz;
    basis[8]  =  0.5462742152960396f * (xx - yy);
    basis[9]  = -0.5900435899266435f * dy * (3.f*xx - yy);
    basis[10] =  2.890611442640554f  * xy * dz;
    basis[11] = -0.4570457994644658f * dy * (4.f*zz - xx - yy);
    basis[12] =  0.3731763325901154f * dz * (2.f*zz - 3.f*xx - 3.f*yy);
    basis[13] = -0.4570457994644658f * dx * (4.f*zz - xx - yy);
    basis[14] =  1.445305721320277f  * dz * (xx - yy);
    basis[15] = -0.5900435899266435f * dx * (xx - 3.f*yy);
    float rgb[3];
    #pragma unroll
    for (int ch = 0; ch < 3; ++ch) {
        float res = 0.f;
        #pragma unroll
        for (int k = 0; k < 16; ++k) res += basis[k] * sh[(size_t)g*48 + k*3 + ch];
        rgb[ch] = fmaxf(res + 0.5f, 0.f);
    }

    float* pk = packed + (size_t)g * 16;
    pk[0] = px;  pk[1] = py;  pk[2] = conA; pk[3] = conB; pk[4] = conC; pk[5] = opac[g];
    pk[6] = rgb[0]; pk[7] = rgb[1]; pk[8] = rgb[2];
    pk[9] = rminx; pk[10] = rminy; pk[11] = rmaxx; pk[12] = rmaxy;
    pk[13] = visible ? 1.f : 0.f; pk[14] = depth; pk[15] = 0.f;

    out_radii[g] = (float)radii;
    out_vis[g]   = visible ? 1.f : 0.f;
    out_xy[2*g]  = px; out_xy[2*g+1] = py;
}

// ---- K2: stable depth rank -> order permutation ----------------------------

__global__ __launch_bounds__(256) void gsplat_order(
    const float* __restrict__ packed, int* __restrict__ order)
{
    __shared__ float s_d[256];
    int g = blockIdx.x * 256 + threadIdx.x;
    float dg = packed[(size_t)g*16 + 14];
    int rank = 0;
    for (int t = 0; t < NCHUNK; ++t) {
        __syncthreads();
        s_d[threadIdx.x] = packed[(size_t)(t*256 + threadIdx.x)*16 + 14];
        __syncthreads();
        int base = t * 256;
        #pragma unroll 8
        for (int j = 0; j < 256; ++j) {
            float dj = s_d[j];
            rank += ((dj < dg) || (dj == dg && (base + j) < g)) ? 1 : 0;
        }
    }
    order[rank] = g;   // unique ranks -> deterministic scatter
}

// ---- K3: tiled rasterizer, double-buffered async staging + WMMA blend -------

__global__ __launch_bounds__(256) void gsplat_raster(
    const float* __restrict__ packed, const int* __restrict__ order,
    const float* __restrict__ bg, float* __restrict__ out_color)
{
    __shared__ float s_raw[2][CHUNKN * 16];     // 2 x 16 KB async-staged chunks
    __shared__ float s_cmp[(CHUNKN + 4) * 12];  // densely compacted splats + zero pad
    __shared__ int   s_red[8];                  // per-wave reduction scratch
    __shared__ float s_col[256 * 4];            // final color exchange

    const int tid  = threadIdx.x;
    const int lane = tid & 31;
    const int wave = tid >> 5;
    const int tx = blockIdx.x, ty = blockIdx.y;
    const int lpx = tid & 15, lpy = tid >> 4;
    const float fpx = (float)(tx * TILEPX + lpx);
    const float fpy = (float)(ty * TILEPX + lpy);
    const float tcol = (float)tx, trow = (float)ty;
    const bool lohalf = lane < 16;

    float Tt = 1.f;
    bool  done = false;
    v8f acc0 = {0.f,0.f,0.f,0.f,0.f,0.f,0.f,0.f};  // pixels lanes 0..15 of wave
    v8f acc1 = {0.f,0.f,0.f,0.f,0.f,0.f,0.f,0.f};  // pixels lanes 16..31 of wave

    // prologue: async-stage chunk 0 into buffer 0
    {
        int g = order[tid];
        const float* src = packed + (size_t)g * 16;
        uint32_t dst = lds_off_u32(&s_raw[0][tid * 16]);
        async_copy_b128(dst +  0, src + 0);
        async_copy_b128(dst + 16, src + 4);
        async_copy_b128(dst + 32, src + 8);
        async_copy_b128(dst + 48, src + 12);
    }

    for (int ch = 0; ch < NCHUNK; ++ch) {
        const int cur = ch & 1;
        wait_async0();          // current buffer's copies (this wave) complete
        __syncthreads();        // all waves' copies visible

        // prefetch next chunk into the other buffer (overlaps blend below)
        if (ch + 1 < NCHUNK) {
            int g = order[(ch + 1) * CHUNKN + tid];
            const float* src = packed + (size_t)g * 16;
            uint32_t dst = lds_off_u32(&s_raw[1 - cur][tid * 16]);
            async_copy_b128(dst +  0, src + 0);
            async_copy_b128(dst + 16, src + 4);
            async_copy_b128(dst + 32, src + 8);
            async_copy_b128(dst + 48, src + 12);
        }

        const float* buf = &s_raw[cur][0];

        // ---- bin: compact tile-intersecting splat *data* (order-preserving) ----
        float rminx = buf[tid*16 +  9], rminy = buf[tid*16 + 10];
        float rmaxx = buf[tid*16 + 11], rmaxy = buf[tid*16 + 12];
        bool pred = (buf[tid*16 + 13] > 0.5f) &&
                    (tcol >= rminx) && (tcol < rmaxx) &&
                    (trow >= rminy) && (trow < rmaxy);
        unsigned bm = __builtin_amdgcn_ballot_w32(pred);
        int before = __builtin_popcount(bm & ((1u << lane) - 1u));
        if (lane == 0) s_red[wave] = __builtin_popcount(bm);
        __syncthreads();
        int waveBase = 0, total = 0;
        #pragma unroll
        for (int w = 0; w < 8; ++w) { int cnt = s_red[w]; if (w < wave) waveBase += cnt; total += cnt; }
        if (pred) {
            float* dc = &s_cmp[(waveBase + before) * 12];
            const float* sc = &buf[tid * 16];
            *(float4*)&dc[0] = *(const float4*)&sc[0];   // px,py,cA,cB
            *(float4*)&dc[4] = *(const float4*)&sc[4];   // cC,op,r,g
            dc[8] = sc[8];                               // b
        }
        int M = total;
        if (tid < 4) {  // zero padding rows -> alpha=0, rgb=0 (branch-free inner loop)
            float* dc = &s_cmp[(M + tid) * 12];
            float4 z = {0.f, 0.f, 0.f, 0.f};
            *(float4*)&dc[0] = z; *(float4*)&dc[4] = z; dc[8] = 0.f;
        }
        __syncthreads();

        int M4 = (M + 3) & ~3;

        // ---- front-to-back blend, 4 splats per WMMA pair, branch-free ----
        for (int j = 0; j < M4; j += 4) {
            float wq[4];
            #pragma unroll
            for (int q = 0; q < 4; ++q) {
                const float* gp = &s_cmp[(j + q) * 12];  // LDS broadcast reads
                float dxp = gp[0] - fpx, dyp = gp[1] - fpy;
                float power = -0.5f*(gp[2]*dxp*dxp + gp[4]*dyp*dyp) - gp[3]*dxp*dyp;
                float G  = __expf(fminf(power, 0.f));
                float al = fminf(0.99f, gp[5] * G);
                al = ((power <= 0.f) && (al >= (1.f/255.f))) ? al : 0.f;
                float test = Tt * (1.f - al);
                bool live = (!done) && (test >= 1e-4f);
                wq[q] = live ? al * Tt : 0.f;
                Tt    = live ? test : Tt;
                done  = done || (test < 1e-4f);
            }
            // A fragments (16x4 f32 layout: lanes0-15 K0,K1 ; lanes16-31 K2,K3)
            int lo = lane & 15;
            float g2 = lane_gather(wq[2], lo);         // lanes>=16 fetch lane-16's K2
            float g3 = lane_gather(wq[3], lo);
            float g0 = lane_gather(wq[0], lane | 16);  // lanes<16 fetch lane+16's K0
            float g1 = lane_gather(wq[1], lane | 16);
            v2f A0 = { lohalf ? wq[0] : g2, lohalf ? wq[1] : g3 };
            v2f A1 = { lohalf ? g0 : wq[2], lohalf ? g1 : wq[3] };
            // B fragment (4x16 f32: rows K0,K1 in lanes0-15; K2,K3 in lanes16-31)
            int   kq   = j + (lohalf ? 0 : 2);
            int   chan = (lo < 3) ? lo : 0;
            float msk  = (lo < 3) ? 1.f : 0.f;
            float b0 = s_cmp[kq      * 12 + 6 + chan] * msk;
            float b1 = s_cmp[(kq+1)  * 12 + 6 + chan] * msk;
            v2f Bf = { b0, b1 };
            acc0 = __builtin_amdgcn_wmma_f32_16x16x4_f32(false, A0, false, Bf,
                                                         (short)0, acc0, false, false);
            acc1 = __builtin_amdgcn_wmma_f32_16x16x4_f32(false, A1, false, Bf,
                                                         (short)0, acc1, false, false);
        }
        __syncthreads();

        // ---- block-uniform early exit when every pixel is saturated ----
        if (lane == 0)
            s_red[wave] = (__builtin_amdgcn_ballot_w32(done) == 0xFFFFFFFFu) ? 1 : 0;
        __syncthreads();
        int nd = 0;
        #pragma unroll
        for (int w = 0; w < 8; ++w) nd += s_red[w];
        if (nd == 8) break;   // outstanding prefetch drains at S_ENDPGM (implicit wait-idle)
    }

    // ---- unpack 16x16 f32 C/D fragments (only N=0..2 columns used) ----
    {
        int n = lane & 15;
        #pragma unroll
        for (int r = 0; r < 8; ++r) {
            if (n < 3) {
                int m = r + (lohalf ? 0 : 8);
                s_col[(wave*32 +      m)*4 + n] = acc0[r];
                s_col[(wave*32 + 16 + m)*4 + n] = acc1[r];
            }
        }
    }
    __syncthreads();
    int px_i = tx * TILEPX + lpx, py_i = ty * TILEPX + lpy;
    #pragma unroll
    for (int c = 0; c < 3; ++c) {
        float col = s_col[tid*4 + c] + Tt * bg[c];
        out_color[(size_t)c*(HH*WW) + py_i*WW + px_i] = col;
    }
}

// ---- host launcher ----------------------------------------------------------

extern "C" void kernel_launch(void* const* d_in, const int* in_sizes, int n_in,
                              void* d_out, int out_size, void* d_ws, size_t ws_size,
                              hipStream_t stream) {
    (void)in_sizes; (void)n_in; (void)out_size; (void)ws_size;
    const float* means  = (const float*)d_in[0];
    const float* sh     = (const float*)d_in[1];
    const float* opac   = (const float*)d_in[2];
    const float* scales = (const float*)d_in[3];
    const float* rots   = (const float*)d_in[4];
    const float* vm     = (const float*)d_in[5];
    const float* pm     = (const float*)d_in[6];
    const float* campos = (const float*)d_in[7];
    const float* bg     = (const float*)d_in[8];

    float* out       = (float*)d_out;
    float* out_color = out;                      // 3*256*256
    float* out_radii = out + 3*HH*WW;            // 4096
    float* out_vis   = out_radii + PTS;          // 4096
    float* out_xy    = out_vis + PTS;            // 8192

    float* packed = (float*)d_ws;                              // PTS*16 floats
    int*   order  = (int*)((char*)d_ws + (size_t)PTS*16*sizeof(float));

    gsplat_preprocess<<<PTS/256, 256, 0, stream>>>(
        means, sh, opac, scales, rots, vm, pm, campos,
        packed, out_radii, out_vis, out_xy);
    gsplat_order<<<PTS/256, 256, 0, stream>>>(packed, order);
    gsplat_raster<<<dim3(WW/TILEPX, HH/TILEPX), 256, 0, stream>>>(
        packed, order, bg, out_color);
}